// SageExtract_84413287236240
// MI455X (gfx1250) — compile-verified
//
#include <hip/hip_runtime.h>

#define N_NODES 50000
#define N_EDGES 800000
#define IN_DIM  128
#define H1      64
#define H2      32
#define OUT_DIM 40

typedef float v2f __attribute__((ext_vector_type(2)));
typedef float v8f __attribute__((ext_vector_type(8)));

// ---------------------------------------------------------------------------
// Scatter layer 1: summed1[dst] += x[src] (128 floats/edge); deg[dst] += 1.
// 32 threads per edge, float4 per thread -> coalesced loads, L2 atomics.
// ---------------------------------------------------------------------------
__global__ __launch_bounds__(256) void scatter1_kernel(
    const float* __restrict__ x, const int* __restrict__ src,
    const int* __restrict__ dst, float* __restrict__ summed,
    float* __restrict__ deg) {
  int t = blockIdx.x * 256 + threadIdx.x;
  if (t >= N_EDGES * 32) return;
  int e = t >> 5;
  int c = t & 31;
  int s = src[e];
  int d = dst[e];
  const float4 v = *(const float4*)(x + s * IN_DIM + c * 4);
  float* o = summed + d * IN_DIM + c * 4;
  atomicAdd(o + 0, v.x);
  atomicAdd(o + 1, v.y);
  atomicAdd(o + 2, v.z);
  atomicAdd(o + 3, v.w);
  if (c == 0) atomicAdd(deg + d, 1.0f);
}

// ---------------------------------------------------------------------------
// Scatter layer 2: summed2[dst] += h1[src] (64 floats/edge).
// ---------------------------------------------------------------------------
__global__ __launch_bounds__(256) void scatter2_kernel(
    const float* __restrict__ h, const int* __restrict__ src,
    const int* __restrict__ dst, float* __restrict__ summed) {
  int t = blockIdx.x * 256 + threadIdx.x;
  if (t >= N_EDGES * 16) return;
  int e = t >> 4;
  int c = t & 15;
  int s = src[e];
  int d = dst[e];
  const float4 v = *(const float4*)(h + s * H1 + c * 4);
  float* o = summed + d * H1 + c * 4;
  atomicAdd(o + 0, v.x);
  atomicAdd(o + 1, v.y);
  atomicAdd(o + 2, v.z);
  atomicAdd(o + 3, v.w);
}

__global__ __launch_bounds__(256) void rdeg_kernel(const float* __restrict__ deg,
                                                   float* __restrict__ rdeg) {
  int n = blockIdx.x * 256 + threadIdx.x;
  if (n < N_NODES) rdeg[n] = 1.0f / fmaxf(deg[n], 1.0f);
}

// ---------------------------------------------------------------------------
// Fused SAGE GEMM tile: out16x16 = relu((sum*rdeg)@Wl^T + xin@Wr^T + bl).
// One wave32 per 16x16 tile using V_WMMA_F32_16X16X4_F32 (fp32, RNE-exact).
// Fragment layouts per CDNA5 ISA 7.12.2:
//   A 16x4 (32b): lane L holds row m=L&15; VGPR j = A[m][2*(L>>4)+j]
//   B 4x16 (32b): lane L holds col n=L&15; VGPR j = B[2*(L>>4)+j][n]
//   C/D 16x16  : VGPR r, lane L -> (row = r + 8*(L>>4), col = L&15)
// ---------------------------------------------------------------------------
template <int K, int NOUT>
__device__ __forceinline__ void sage_tile(
    const float* __restrict__ sum, const float* __restrict__ xin,
    const float* __restrict__ rdeg, const float* __restrict__ Wl,
    const float* __restrict__ bl, const float* __restrict__ Wr,
    float* __restrict__ out, int rowTile, int nTile, int lane) {
  const int m   = rowTile * 16 + (lane & 15);
  const int kHi = (lane >> 4) * 2;
  const int n   = nTile * 16 + (lane & 15);
  const float rd = rdeg[m];
  const float* As = sum + m * K;
  const float* Ax = xin + m * K;
  const float* Bl = Wl + n * K;  // row n of W == column n of W^T
  const float* Br = Wr + n * K;
  v8f acc = {};
  for (int kb = 0; kb < K; kb += 4) {
    const int k0 = kb + kHi;
    v2f am = {As[k0] * rd, As[k0 + 1] * rd};
    v2f bm = {Bl[k0], Bl[k0 + 1]};
    acc = __builtin_amdgcn_wmma_f32_16x16x4_f32(false, am, false, bm,
                                                (short)0, acc, false, false);
    v2f ax = {Ax[k0], Ax[k0 + 1]};
    v2f br = {Br[k0], Br[k0 + 1]};
    acc = __builtin_amdgcn_wmma_f32_16x16x4_f32(false, ax, false, br,
                                                (short)0, acc, false, false);
  }
  const float bias = bl[n];
  const int rowBase = rowTile * 16 + 8 * (lane >> 4);
#pragma unroll
  for (int r = 0; r < 8; ++r) {
    out[(rowBase + r) * NOUT + n] = fmaxf(acc[r] + bias, 0.0f);
  }
}

// Layer 1: [N,128] -> [N,64]; 4 waves/block cover 4 column tiles.
__global__ __launch_bounds__(128) void gemm1_kernel(
    const float* __restrict__ summed, const float* __restrict__ x,
    const float* __restrict__ rdeg, const float* __restrict__ Wl,
    const float* __restrict__ bl, const float* __restrict__ Wr,
    float* __restrict__ h1) {
  sage_tile<IN_DIM, H1>(summed, x, rdeg, Wl, bl, Wr, h1, blockIdx.x,
                        threadIdx.x >> 5, threadIdx.x & 31);
}

// Layer 2: [N,64] -> [N,32]; 2 waves/block; writes embedding into d_out.
__global__ __launch_bounds__(64) void gemm2_kernel(
    const float* __restrict__ summed, const float* __restrict__ h1,
    const float* __restrict__ rdeg, const float* __restrict__ Wl,
    const float* __restrict__ bl, const float* __restrict__ Wr,
    float* __restrict__ emb) {
  sage_tile<H1, H2>(summed, h1, rdeg, Wl, bl, Wr, emb, blockIdx.x,
                    threadIdx.x >> 5, threadIdx.x & 31);
}

// Final FC: logits = h2 @ Wfc^T + bfc  ([N,32] x [32,40]); 40 cols padded to
// 48 (3 waves). Out-of-range columns get zeroed B fragments via arithmetic
// mask so EXEC stays all-1s through every WMMA; stores are guarded after.
__global__ __launch_bounds__(96) void fc_kernel(
    const float* __restrict__ h2, const float* __restrict__ Wfc,
    const float* __restrict__ bfc, float* __restrict__ logits) {
  const int lane = threadIdx.x & 31;
  const int nTile = threadIdx.x >> 5;
  const int m   = blockIdx.x * 16 + (lane & 15);
  const int kHi = (lane >> 4) * 2;
  const int n   = nTile * 16 + (lane & 15);
  const int nc  = n < OUT_DIM ? n : OUT_DIM - 1;  // clamp address
  const float mask = n < OUT_DIM ? 1.0f : 0.0f;
  const float* A = h2 + m * H2;
  const float* B = Wfc + nc * H2;
  v8f acc = {};
#pragma unroll
  for (int kb = 0; kb < H2; kb += 4) {
    const int k0 = kb + kHi;
    v2f a = {A[k0], A[k0 + 1]};
    v2f b = {B[k0] * mask, B[k0 + 1] * mask};
    acc = __builtin_amdgcn_wmma_f32_16x16x4_f32(false, a, false, b,
                                                (short)0, acc, false, false);
  }
  if (n < OUT_DIM) {
    const float bias = bfc[n];
    const int rowBase = blockIdx.x * 16 + 8 * (lane >> 4);
#pragma unroll
    for (int r = 0; r < 8; ++r) {
      logits[(rowBase + r) * OUT_DIM + n] = acc[r] + bias;
    }
  }
}

extern "C" void kernel_launch(void* const* d_in, const int* in_sizes, int n_in,
                              void* d_out, int out_size, void* d_ws,
                              size_t ws_size, hipStream_t stream) {
  (void)in_sizes; (void)n_in; (void)out_size; (void)ws_size;

  const float* x   = (const float*)d_in[0];
  const int*   ei  = (const int*)d_in[1];
  const int*   src = ei;
  const int*   dst = ei + N_EDGES;
  const float* Wl1 = (const float*)d_in[2];
  const float* bl1 = (const float*)d_in[3];
  const float* Wr1 = (const float*)d_in[4];
  const float* Wl2 = (const float*)d_in[5];
  const float* bl2 = (const float*)d_in[6];
  const float* Wr2 = (const float*)d_in[7];
  const float* Wfc = (const float*)d_in[8];
  const float* bfc = (const float*)d_in[9];

  // Workspace layout (floats). Zeroed region must be contiguous & first.
  float* ws      = (float*)d_ws;
  float* deg     = ws;                                  // N
  float* summed1 = deg + N_NODES;                       // N*128
  float* summed2 = summed1 + (size_t)N_NODES * IN_DIM;  // N*64
  float* rdeg    = summed2 + (size_t)N_NODES * H1;      // N
  float* h1      = rdeg + N_NODES;                      // N*64

  float* emb    = (float*)d_out;                 // [N,32] embedding
  float* logits = emb + (size_t)N_NODES * H2;    // [N,40] logits

  // Zero accumulation buffers every call (harness does not re-zero d_ws).
  const size_t zeroBytes = sizeof(float) * (size_t)N_NODES * (1 + IN_DIM + H1);
  hipMemsetAsync(d_ws, 0, zeroBytes, stream);

  {
    const int T = N_EDGES * 32;
    scatter1_kernel<<<(T + 255) / 256, 256, 0, stream>>>(x, src, dst, summed1,
                                                         deg);
  }
  rdeg_kernel<<<(N_NODES + 255) / 256, 256, 0, stream>>>(deg, rdeg);

  gemm1_kernel<<<N_NODES / 16, 128, 0, stream>>>(summed1, x, rdeg, Wl1, bl1,
                                                 Wr1, h1);
  {
    const int T = N_EDGES * 16;
    scatter2_kernel<<<(T + 255) / 256, 256, 0, stream>>>(h1, src, dst, summed2);
  }
  gemm2_kernel<<<N_NODES / 16, 64, 0, stream>>>(summed2, h1, rdeg, Wl2, bl2,
                                                Wr2, emb);
  fc_kernel<<<N_NODES / 16, 96, 0, stream>>>(emb, Wfc, bfc, logits);
}